// model2_variant1_2104533975375
// MI455X (gfx1250) — compile-verified
//
#include <hip/hip_runtime.h>

typedef __attribute__((ext_vector_type(2))) float v2f;
typedef __attribute__((ext_vector_type(8))) float v8f;

#define ATOM_ADDF(p, v) __hip_atomic_fetch_add((p), (v), __ATOMIC_RELAXED, __HIP_MEMORY_SCOPE_AGENT)
#define ATOM_ADDI(p, v) __hip_atomic_fetch_add((p), (v), __ATOMIC_RELAXED, __HIP_MEMORY_SCOPE_AGENT)

constexpr int N_TOTAL   = 16384;
constexpr int N_EDGES   = 524288;
constexpr int BATCH     = 8;
constexpr int LATENT    = 512;
constexpr int NUM_NODES = 2048;
constexpr int ENC_K     = NUM_NODES * 256;   // 524288
constexpr int GK        = 512;               // K of all three GCN GEMMs
constexpr int NF        = N_TOTAL * 512;

__device__ __forceinline__ v8f wmma4(v2f a, v2f b, v8f c) {
  // V_WMMA_F32_16X16X4_F32 : D = A(16x4 f32) * B(4x16 f32) + C(16x16 f32)
  return __builtin_amdgcn_wmma_f32_16x16x4_f32(false, a, false, b, (short)0, c, false, false);
}

// ---------------------------------------------------------------------------
// C[M,N] = A[M,K=512] @ B[K,N].  8 waves/block; all waves share one 32-row
// A block staged through LDS (K-sliced, padded stride 260 -> conflict-free
// ds_load_b64 fragment reads).  Each wave owns a 32x32 C tile.
// ---------------------------------------------------------------------------
__global__ __launch_bounds__(256) void gemm_wmma_f32(
    const float* __restrict__ A, const float* __restrict__ B, float* __restrict__ C,
    int N) {
  constexpr int LDA = 260;                      // 256 + 4 pad (floats)
  __shared__ float As[32 * LDA];                // 33,280 B

  const int tid  = threadIdx.x;
  const int lane = tid & 31;
  const int wave = tid >> 5;
  const int tiles_n = N >> 5;
  const int t    = blockIdx.x * 8 + wave;
  const int m0b  = ((blockIdx.x * 8) / tiles_n) << 5;   // block-uniform M row
  const int n0   = (t % tiles_n) << 5;
  const int half = lane >> 4;
  const int l15  = lane & 15;

  const float* Ablk = A + (size_t)m0b * GK;
  const float* bb0  = B + (size_t)(2 * half) * N + n0 + l15;
  const float* bb1  = bb0 + 16;
  const float* a0s  = As + l15 * LDA + 2 * half;
  const float* a1s  = a0s + 16 * LDA;

  v8f acc00 = {}, acc01 = {}, acc10 = {}, acc11 = {};
  for (int kb = 0; kb < GK; kb += 256) {
    // stage A[m0b..m0b+31, kb..kb+255] : 32 rows x 64 float4
    for (int c = tid; c < 32 * 64; c += 256) {
      int r = c >> 6, col = (c & 63) << 2;
      *(float4*)&As[r * LDA + col] = *(const float4*)(Ablk + (size_t)r * GK + kb + col);
    }
    __syncthreads();
    for (int k = 0; k < 256; k += 4) {
      v2f a0 = *(const v2f*)(a0s + k);          // ds_load_b64
      v2f a1 = *(const v2f*)(a1s + k);
      size_t ro = (size_t)(kb + k) * N;
      v2f b0, b1;
      b0.x = bb0[ro]; b0.y = bb0[ro + N];
      b1.x = bb1[ro]; b1.y = bb1[ro + N];
      acc00 = wmma4(a0, b0, acc00);
      acc01 = wmma4(a0, b1, acc01);
      acc10 = wmma4(a1, b0, acc10);
      acc11 = wmma4(a1, b1, acc11);
    }
    __syncthreads();
  }
  for (int v = 0; v < 8; ++v) {
    int mr = m0b + v + 8 * half;
    C[(size_t)mr * N + n0 + l15]             = acc00[v];
    C[(size_t)mr * N + n0 + 16 + l15]        = acc01[v];
    C[(size_t)(mr + 16) * N + n0 + l15]      = acc10[v];
    C[(size_t)(mr + 16) * N + n0 + 16 + l15] = acc11[v];
  }
}

// ---------------------------------------------------------------------------
// CSR construction (per-call, cheap: ~1M int atomics + one small scan)
// ---------------------------------------------------------------------------
__global__ void zero_i32(int* __restrict__ p, int n) {
  int i = blockIdx.x * blockDim.x + threadIdx.x;
  if (i < n) p[i] = 0;
}

__global__ void count_edges(const long long* __restrict__ ei, int* __restrict__ cnt) {
  int e = blockIdx.x * blockDim.x + threadIdx.x;
  if (e < N_EDGES) ATOM_ADDI(&cnt[(int)ei[N_EDGES + e]], 1);
}

// dinv[i] = rsqrt(in_degree + 1 self loop)
__global__ void make_dinv(const int* __restrict__ cnt, float* __restrict__ d) {
  int i = blockIdx.x * blockDim.x + threadIdx.x;
  if (i < N_TOTAL) d[i] = rsqrtf((float)cnt[i] + 1.0f);
}

// exclusive prefix sum of cnt[0..16383] -> off[0..16384]; single block
__global__ __launch_bounds__(256) void scan_offsets(const int* __restrict__ cnt,
                                                    int* __restrict__ off) {
  __shared__ int part[256];
  int tid = threadIdx.x;
  int base = tid * 64;
  int s = 0;
  for (int i = 0; i < 64; ++i) s += cnt[base + i];
  part[tid] = s;
  __syncthreads();
  int pre = 0;
  for (int i = 0; i < tid; ++i) pre += part[i];
  int run = pre;
  for (int i = 0; i < 64; ++i) { off[base + i] = run; run += cnt[base + i]; }
  if (tid == 255) off[N_TOTAL] = run;
}

__global__ void copy_i32(const int* __restrict__ src, int* __restrict__ dst, int n) {
  int i = blockIdx.x * blockDim.x + threadIdx.x;
  if (i < n) dst[i] = src[i];
}

__global__ void build_csr(const long long* __restrict__ ei, int* __restrict__ pos,
                          int* __restrict__ esrc) {
  int e = blockIdx.x * blockDim.x + threadIdx.x;
  if (e < N_EDGES) {
    int c = (int)ei[N_EDGES + e];
    int idx = ATOM_ADDI(&pos[c], 1);
    esrc[idx] = (int)ei[e];
  }
}

// ---------------------------------------------------------------------------
// Gather aggregation (no float atomics):
// out[c] = [relu]( dinv[c] * sum_{s in N(c)} dinv[s]*Hw[s]  +  Hw[c]*dinv[c]^2 + b )
// One block per destination node; threads own feature columns.
// ---------------------------------------------------------------------------
__global__ __launch_bounds__(256) void gather_nodes(
    const float* __restrict__ Hw, const int* __restrict__ off, const int* __restrict__ esrc,
    const float* __restrict__ dinv, const float* __restrict__ bias,
    float* __restrict__ out, int F, int relu) {
  int c = blockIdx.x;
  int beg = off[c], end = off[c + 1];
  float dc = dinv[c];
  int f0 = threadIdx.x;
  float acc0 = 0.0f, acc1 = 0.0f;
  for (int j = beg; j < end; ++j) {
    int s = esrc[j];
    float w = dinv[s];
    const float* hp = Hw + (size_t)s * F;
    acc0 = fmaf(w, hp[f0], acc0);
    if (F > 256) acc1 = fmaf(w, hp[f0 + 256], acc1);
  }
  size_t base = (size_t)c * F;
  float v0 = dc * acc0 + Hw[base + f0] * dc * dc + bias[f0];
  if (relu) v0 = fmaxf(v0, 0.0f);
  out[base + f0] = v0;
  if (F > 256) {
    float v1 = dc * acc1 + Hw[base + f0 + 256] * dc * dc + bias[f0 + 256];
    if (relu) v1 = fmaxf(v1, 0.0f);
    out[base + f0 + 256] = v1;
  }
}

// ---------------------------------------------------------------------------
// Encoder: z[8,512] = h[8,524288] @ We[524288,512] + be.  Split-K: wave job =
// (n-tile of 32 cols, K-chunk of 2048); all 8 waves of a block share one
// K-chunk, whose 16 H rows (8 real + 8 in-bounds pad) are staged in LDS.
// We (1.07 GB, the HBM-bound term) is streamed exactly once + prefetched.
// ---------------------------------------------------------------------------
__global__ __launch_bounds__(256) void encoder_splitk(
    const float* __restrict__ H, const float* __restrict__ We, float* __restrict__ Z) {
  constexpr int LDH = 516;                      // 512 + 4 pad
  __shared__ float Hs[16 * LDH];                // 33,024 B

  const int tid  = threadIdx.x;
  const int lane = tid & 31;
  const int wave = tid >> 5;
  const int job  = blockIdx.x * 8 + wave;       // 4096 jobs
  const int n0   = (job & 15) << 5;             // 16 n-tiles
  const int k0   = ((blockIdx.x * 8) >> 4) << 11;  // block-uniform chunk * 2048
  const int half = lane >> 4;
  const int l15  = lane & 15;

  const float* bb0 = We + (size_t)(2 * half) * LATENT + n0 + l15;
  const float* bb1 = bb0 + 16;
  const float* a0s = Hs + l15 * LDH + 2 * half;

  v8f acc0 = {}, acc1 = {};
  for (int kb = 0; kb < 2048; kb += 512) {
    // stage H[0..15, k0+kb .. +511] : 16 rows x 128 float4 (rows 8-15 = pad)
    for (int c = tid; c < 16 * 128; c += 256) {
      int r = c >> 7, col = (c & 127) << 2;
      *(float4*)&Hs[r * LDH + col] = *(const float4*)(H + (size_t)r * ENC_K + k0 + kb + col);
    }
    __syncthreads();
    for (int k = 0; k < 512; k += 4) {
      v2f a = *(const v2f*)(a0s + k);           // ds_load_b64
      size_t ro = (size_t)(k0 + kb + k) * LATENT;
      v2f b0, b1;
      b0.x = bb0[ro]; b0.y = bb0[ro + LATENT];
      b1.x = bb1[ro]; b1.y = bb1[ro + LATENT];
      __builtin_prefetch(bb0 + ro + (size_t)32 * LATENT, 0, 1);  // global_prefetch
      acc0 = wmma4(a, b0, acc0);
      acc1 = wmma4(a, b1, acc1);
    }
    __syncthreads();
  }
  if (half == 0) {                              // only real rows 0..7 stored
    for (int v = 0; v < 8; ++v) {
      ATOM_ADDF(&Z[(size_t)v * LATENT + n0 + l15],      acc0[v]);
      ATOM_ADDF(&Z[(size_t)v * LATENT + n0 + 16 + l15], acc1[v]);
    }
  }
}

__global__ void init_z(float* __restrict__ z, const float* __restrict__ be) {
  int idx = blockIdx.x * blockDim.x + threadIdx.x;
  if (idx < BATCH * LATENT) z[idx] = be[idx & (LATENT - 1)];
}

// out[8,2048] = z[8,512] @ Wd[512,2048] + bd  (tiny; plain VALU dot products)
__global__ __launch_bounds__(256) void decoder_kernel(
    const float* __restrict__ Z, const float* __restrict__ Wd,
    const float* __restrict__ bd, float* __restrict__ out) {
  int idx = blockIdx.x * blockDim.x + threadIdx.x;
  if (idx >= BATCH * NUM_NODES) return;
  int n = idx & (NUM_NODES - 1);
  int m = idx >> 11;
  const float* zp = Z + (size_t)m * LATENT;
  const float* wp = Wd + n;
  float s = bd[n];
  for (int k = 0; k < LATENT; ++k)
    s = fmaf(zp[k], wp[(size_t)k * NUM_NODES], s);
  out[idx] = s;
}

// ---------------------------------------------------------------------------
extern "C" void kernel_launch(void* const* d_in, const int* in_sizes, int n_in,
                              void* d_out, int out_size, void* d_ws, size_t ws_size,
                              hipStream_t stream) {
  (void)in_sizes; (void)n_in; (void)out_size; (void)ws_size;
  const float*     x  = (const float*)d_in[0];
  const long long* ei = (const long long*)d_in[1];
  const float* W1 = (const float*)d_in[4];
  const float* b1 = (const float*)d_in[5];
  const float* W2 = (const float*)d_in[6];
  const float* b2 = (const float*)d_in[7];
  const float* W3 = (const float*)d_in[8];
  const float* b3 = (const float*)d_in[9];
  const float* We = (const float*)d_in[10];
  const float* be = (const float*)d_in[11];
  const float* Wd = (const float*)d_in[12];
  const float* bd = (const float*)d_in[13];
  float* out = (float*)d_out;

  float* ws   = (float*)d_ws;
  float* dinv = ws;                            // 16384
  float* bufA = ws + 16384;                    // Hw scratch, 16384*512
  float* bufB = bufA + (size_t)NF;             // 16384*512
  float* bufC = bufB + (size_t)NF;             // 16384*512
  float* z    = bufC + (size_t)NF;             // 8*512
  int*   cnt  = (int*)(z + BATCH * LATENT);    // 16384
  int*   off  = cnt + N_TOTAL;                 // 16385
  int*   pos  = off + N_TOTAL + 1;             // 16384
  int*   esrc = pos + N_TOTAL;                 // 524288

  // ---- CSR + symmetric-norm degree ----
  zero_i32   <<<N_TOTAL / 256, 256, 0, stream>>>(cnt, N_TOTAL);
  count_edges<<<N_EDGES / 256, 256, 0, stream>>>(ei, cnt);
  make_dinv  <<<N_TOTAL / 256, 256, 0, stream>>>(cnt, dinv);
  scan_offsets<<<1, 256, 0, stream>>>(cnt, off);
  copy_i32   <<<N_TOTAL / 256, 256, 0, stream>>>(off, pos, N_TOTAL);
  build_csr  <<<N_EDGES / 256, 256, 0, stream>>>(ei, pos, esrc);

  // ---- layer 1: h1 = relu(Agg(x@W1) + b1) -> bufB ----
  gemm_wmma_f32<<<(N_TOTAL / 32) * (512 / 32) / 8, 256, 0, stream>>>(x, W1, bufA, 512);
  gather_nodes <<<N_TOTAL, 256, 0, stream>>>(bufA, off, esrc, dinv, b1, bufB, 512, 1);

  // ---- layer 2: h2 = relu(Agg(h1@W2) + b2) -> bufC ----
  gemm_wmma_f32<<<(N_TOTAL / 32) * (512 / 32) / 8, 256, 0, stream>>>(bufB, W2, bufA, 512);
  gather_nodes <<<N_TOTAL, 256, 0, stream>>>(bufA, off, esrc, dinv, b2, bufC, 512, 1);

  // ---- layer 3: h3 = Agg(h2@W3) + b3 -> bufB (no relu) ----
  gemm_wmma_f32<<<(N_TOTAL / 32) * (256 / 32) / 8, 256, 0, stream>>>(bufC, W3, bufA, 256);
  gather_nodes <<<N_TOTAL, 256, 0, stream>>>(bufA, off, esrc, dinv, b3, bufB, 256, 0);

  // ---- encoder: z = h3.reshape(8,524288) @ We + be (split-K, We read once) ----
  init_z<<<(BATCH * LATENT) / 256, 256, 0, stream>>>(z, be);
  encoder_splitk<<<(16 * 256) / 8, 256, 0, stream>>>(bufB, We, z);

  // ---- decoder: out = z @ Wd + bd ----
  decoder_kernel<<<(BATCH * NUM_NODES) / 256, 256, 0, stream>>>(z, Wd, bd, out);
}